// HamiltonianPositionProducer_46918222741665
// MI455X (gfx1250) — compile-verified
//
#include <hip/hip_runtime.h>
#include <math.h>

#define N_MOL 128
#define N_V   2048
#define N_E   8192
#define N_DIM 128
#define E_DIM 64
#define PQ    3
#define HID   64
#define TAUF  0.25f
#define N_LAYERS 4
#define UEV_DIM 320
#define B1N   80      // 64 hidden cols + 1 e-encoder col + 15 pad
#define KCHUNK 64

typedef __attribute__((ext_vector_type(2))) float v2f;
typedef __attribute__((ext_vector_type(8))) float v8f;

// scalar slot layout in ws
#define S_POT  0   // [0..3] pot per layer
#define S_KIN  4   // [4..7]
#define S_D    8   // [8..11]
#define S_S2   12  // [12..15]
#define S_SL   16  // accumulated s_loss
#define S_CL   17  // accumulated c_loss

// ---------------- index extraction from one-hot matrices ----------------
__global__ void k_extract_edges(const float* __restrict__ vew1,
                                const float* __restrict__ vew2,
                                int* __restrict__ src, int* __restrict__ dst) {
    int j = blockIdx.x * blockDim.x + threadIdx.x;
    if (j >= N_E) return;
    int s = 0, d = 0;
    for (int i = 0; i < N_V; ++i) {
        if (vew1[(size_t)i * N_E + j] != 0.f) s = i;
        if (vew2[(size_t)i * N_E + j] != 0.f) d = i;
    }
    src[j] = s; dst[j] = d;
}

__global__ void k_extract_mol(const float* __restrict__ mvw, int* __restrict__ mol) {
    int v = blockIdx.x * blockDim.x + threadIdx.x;
    if (v >= N_V) return;
    int m = 0;
    for (int i = 0; i < N_MOL; ++i)
        if (mvw[(size_t)i * N_V + v] != 0.f) m = i;
    mol[v] = m;
}

// ---------------- build uev [N_E, 320] ----------------
__global__ void k_build_uev(const float* __restrict__ vf, const float* __restrict__ ef,
                            const int* __restrict__ src, const int* __restrict__ dst,
                            float* __restrict__ uev) {
    int idx = blockIdx.x * blockDim.x + threadIdx.x;
    if (idx >= N_E * UEV_DIM) return;
    int j = idx / UEV_DIM;
    int k = idx - j * UEV_DIM;
    float val;
    if (k < N_DIM)              val = vf[(size_t)src[j] * N_DIM + k];
    else if (k < N_DIM + E_DIM) val = ef[(size_t)j * E_DIM + (k - N_DIM)];
    else                        val = vf[(size_t)dst[j] * N_DIM + (k - N_DIM - E_DIM)];
    uev[idx] = val;
}

// ---------------- composite B1 [320, 80] and bias1 [80] ----------------
__global__ void k_build_B1(const float* __restrict__ W_h1, const float* __restrict__ b_h1,
                           const float* __restrict__ W_e, const float* __restrict__ b_e,
                           float* __restrict__ B1, float* __restrict__ bias1) {
    int idx = blockIdx.x * blockDim.x + threadIdx.x;
    if (idx >= UEV_DIM * B1N) return;
    int r = idx / B1N, c = idx - r * B1N;
    float v = 0.f;
    if (c < HID)       v = W_h1[r * HID + c];   // W_h1 rows 0..319
    else if (c == HID) v = W_e[r];              // W_e [320,1]
    B1[idx] = v;
    if (idx < B1N)
        bias1[idx] = (idx < HID) ? b_h1[idx] : (idx == HID ? b_e[0] : 0.f);
}

// ---------------- generic f32 WMMA GEMM: C[M,N] = A[M,K] @ B[K,N] + bias ----------------
// K, N compile-time so the accumulator tile array is fully register-resident.
// blockDim = 128 (4 waves); each wave: one 16-row tile x all N cols.
// Requires N%16==0, N<=80, K%KCHUNK==0, M%64==0.
// B staged through LDS in 64-row chunks, pre-paired as float2 {B[2t][c], B[2t+1][c]}
// so each lane's fragment read is a single aligned ds_load_b64 into an even VGPR pair.
template <int K, int N>
__global__ void k_wmma_gemm_f32(const float* __restrict__ A, const float* __restrict__ B,
                                const float* __restrict__ bias, float* __restrict__ C,
                                int M) {
    constexpr int NT = N / 16;                 // number of 16-col tiles (<=5)
    __shared__ __align__(8) float sB[KCHUNK * N];   // logical: float2[(KCHUNK/2)][N]
    const int wave = threadIdx.x >> 5;
    const int lane = threadIdx.x & 31;
    const int half = lane >> 4;                // 0: lanes 0-15, 1: lanes 16-31
    const int l16  = lane & 15;
    const int row0 = (blockIdx.x * 4 + wave) * 16;

    v8f acc[NT];
#pragma unroll
    for (int n = 0; n < NT; ++n)
        acc[n] = (v8f){0.f,0.f,0.f,0.f,0.f,0.f,0.f,0.f};

    for (int k0 = 0; k0 < K; k0 += KCHUNK) {
        // cooperative staging with row-pair swizzle:
        //   element (r,c) of the chunk -> sB[((r>>1)*N + c)*2 + (r&1)]
        const float* __restrict__ Bc = B + (size_t)k0 * N;
#pragma unroll 4
        for (int idx = threadIdx.x; idx < KCHUNK * N; idx += 128) {
            int r = idx / N, c = idx - r * N;
            sB[((r >> 1) * N + c) * 2 + (r & 1)] = Bc[idx];
        }
        __syncthreads();

#pragma unroll 4
        for (int kk = 0; kk < KCHUNK; kk += 4) {
            // A fragment (16x4 f32): lanes 0-15 -> K = kk+0,kk+1 ; lanes 16-31 -> kk+2,kk+3
            v2f a = *(const v2f*)(A + (size_t)(row0 + l16) * K + (k0 + kk + 2 * half));
            const int t = (kk >> 1) + half;    // row pair (kk+2*half, kk+2*half+1)
#pragma unroll
            for (int n = 0; n < NT; ++n) {
                v2f b = *(const v2f*)&sB[(t * N + n * 16 + l16) * 2];
                acc[n] = __builtin_amdgcn_wmma_f32_16x16x4_f32(
                    false, a, false, b, (short)0, acc[n], false, false);
            }
        }
        __syncthreads();
    }

    // D layout: element r of acc -> row = row0 + r + 8*half, col = n*16 + l16
#pragma unroll
    for (int n = 0; n < NT; ++n) {
        int cc = n * 16 + l16;
        float bv = bias ? bias[cc] : 0.f;
#pragma unroll
        for (int r = 0; r < 8; ++r) {
            int rr = row0 + r + 8 * half;
            C[(size_t)rr * N + cc] = acc[n][r] + bv;
        }
    }
}

// ---------------- p0 / q0(part) / vq per vertex ----------------
__global__ void k_vertex_init(const float* __restrict__ vf,
                              const float* __restrict__ W_qe, const float* __restrict__ W_q,
                              const float* __restrict__ W_p,
                              float* __restrict__ vq, float* __restrict__ q,
                              float* __restrict__ p) {
    int u = blockIdx.x * blockDim.x + threadIdx.x;
    if (u >= N_V) return;
    const float* row = vf + (size_t)u * N_DIM;
    float aqe[PQ] = {0,0,0}, aq[PQ] = {0,0,0}, ap[PQ] = {0,0,0};
    for (int k = 0; k < N_DIM; ++k) {
        float x = row[k];
        for (int c = 0; c < PQ; ++c) {
            aqe[c] += x * W_qe[k * PQ + c];
            aq[c]  += x * W_q[k * PQ + c];
            ap[c]  += x * W_p[k * PQ + c];
        }
    }
    for (int c = 0; c < PQ; ++c) {
        vq[u * PQ + c] = aqe[c];
        q[u * PQ + c]  = aq[c];   // v@W_q part of q0; edge scatter adds the rest
        p[u * PQ + c]  = ap[c];
    }
}

// q0 += e @ (v @ W_qe): per-edge scatter of w_j * vq[dst] into row src
__global__ void k_edge_scatter_q0(const float* __restrict__ HB,
                                  const int* __restrict__ src, const int* __restrict__ dst,
                                  const float* __restrict__ vq, float* __restrict__ q) {
    int j = blockIdx.x * blockDim.x + threadIdx.x;
    if (j >= N_E) return;
    float logit = HB[(size_t)j * B1N + HID];      // uev@W_e + b_e (col 64)
    float w = 1.f / (1.f + __expf(-logit));
    int s = src[j], d = dst[j];
    for (int c = 0; c < PQ; ++c)
        atomicAdd(&q[s * PQ + c], w * vq[d * PQ + c]);
}

__global__ void k_zero(float* __restrict__ ptr, int n) {
    int i = blockIdx.x * blockDim.x + threadIdx.x;
    if (i < n) ptr[i] = 0.f;
}

// ---------------- per-layer edge pass: pot, g_j, scatter dHdq ----------------
__global__ void k_edge_layer(const float* __restrict__ HB,
                             const int* __restrict__ src, const int* __restrict__ dst,
                             const float* __restrict__ q,
                             const float* __restrict__ W_h1, const float* __restrict__ W_h2,
                             float* __restrict__ dHdq, float* __restrict__ scal, int layer) {
    int j = blockIdx.x * blockDim.x + threadIdx.x;
    float pot_local = 0.f;
    if (j < N_E) {
        int s = src[j], d = dst[j];
        float d0 = q[s*PQ+0] - q[d*PQ+0];
        float d1 = q[s*PQ+1] - q[d*PQ+1];
        float d2 = q[s*PQ+2] - q[d*PQ+2];
        float dist2 = d0*d0 + d1*d1 + d2*d2;
        const float* hb   = HB + (size_t)j * B1N;
        const float* wrow = W_h1 + (size_t)UEV_DIM * HID;   // W_h1 row 320
        float g = 0.f;
        for (int k = 0; k < HID; ++k) {
            float a  = hb[k] + dist2 * wrow[k];
            float t  = tanhf(a);
            float w2 = W_h2[k];
            pot_local += t * w2;
            g += (1.f - t * t) * wrow[k] * w2;
        }
        float coef = 2.f * g;
        atomicAdd(&dHdq[s*PQ+0],  coef * d0);
        atomicAdd(&dHdq[s*PQ+1],  coef * d1);
        atomicAdd(&dHdq[s*PQ+2],  coef * d2);
        atomicAdd(&dHdq[d*PQ+0], -coef * d0);
        atomicAdd(&dHdq[d*PQ+1], -coef * d1);
        atomicAdd(&dHdq[d*PQ+2], -coef * d2);
    }
    __shared__ float red[256];
    red[threadIdx.x] = pot_local;
    __syncthreads();
    for (int st = blockDim.x >> 1; st > 0; st >>= 1) {
        if ((int)threadIdx.x < st) red[threadIdx.x] += red[threadIdx.x + st];
        __syncthreads();
    }
    if (threadIdx.x == 0) atomicAdd(&scal[S_POT + layer], red[0]);
}

// ---------------- per-layer vertex pass: dissipation, kin, integrate ----------------
__global__ void k_vertex_layer(const float* __restrict__ DB, const float* __restrict__ mass,
                               const float* __restrict__ W_d1, const float* __restrict__ W_d2,
                               const int* __restrict__ mol,
                               float* __restrict__ p, float* __restrict__ q,
                               const float* __restrict__ dHdq,
                               float* __restrict__ molAcc, float* __restrict__ scal, int layer) {
    int u = blockIdx.x * blockDim.x + threadIdx.x;
    float kin_l = 0.f, d_l = 0.f, s2_l = 0.f;
    if (u < N_V) {
        float pu[PQ], qu[PQ];
        for (int c = 0; c < PQ; ++c) { pu[c] = p[u*PQ+c]; qu[c] = q[u*PQ+c]; }
        const float* db = DB + (size_t)u * HID;
        float ddp[PQ] = {0,0,0};
        for (int k = 0; k < HID; ++k) {
            float a = db[k];
            for (int c = 0; c < PQ; ++c)
                a += pu[c] * W_d1[(N_DIM + c) * HID + k]
                   + qu[c] * W_d1[(N_DIM + PQ + c) * HID + k];
            if (a > 0.f) {
                float w2 = W_d2[k];
                d_l += a * w2;
                for (int c = 0; c < PQ; ++c)
                    ddp[c] += w2 * W_d1[(N_DIM + c) * HID + k];
            }
        }
        float m = mass[u];
        kin_l = 0.5f * m * (pu[0]*pu[0] + pu[1]*pu[1] + pu[2]*pu[2]);
        int mm = mol[u];
        for (int c = 0; c < PQ; ++c) {
            float dqv = m * pu[c];                       // dHdp
            float dp  = -dHdq[u*PQ+c] - ddp[c];
            float sd  = dqv - pu[c];
            s2_l += sd * sd;
            float dstep = TAUF * dp;                     // p_new - p
            atomicAdd(&molAcc[mm * PQ + c], dstep);
            p[u*PQ+c] = pu[c] + dstep;
            q[u*PQ+c] = qu[c] + TAUF * dqv;
        }
    }
    __shared__ float rk[256], rd[256], rs[256];
    rk[threadIdx.x] = kin_l; rd[threadIdx.x] = d_l; rs[threadIdx.x] = s2_l;
    __syncthreads();
    for (int st = blockDim.x >> 1; st > 0; st >>= 1) {
        if ((int)threadIdx.x < st) {
            rk[threadIdx.x] += rk[threadIdx.x + st];
            rd[threadIdx.x] += rd[threadIdx.x + st];
            rs[threadIdx.x] += rs[threadIdx.x + st];
        }
        __syncthreads();
    }
    if (threadIdx.x == 0) {
        atomicAdd(&scal[S_KIN + layer], rk[0]);
        atomicAdd(&scal[S_D   + layer], rd[0]);
        atomicAdd(&scal[S_S2  + layer], rs[0]);
    }
}

// ---------------- per-layer loss finalize (1 block) ----------------
__global__ void k_finalize_layer(const float* __restrict__ molAcc, float* __restrict__ scal,
                                 int layer) {
    __shared__ float red[128];
    float v = 0.f;
    for (int i = threadIdx.x; i < N_MOL * PQ; i += blockDim.x)
        v += molAcc[i] * molAcc[i];
    red[threadIdx.x] = v;
    __syncthreads();
    for (int st = blockDim.x >> 1; st > 0; st >>= 1) {
        if ((int)threadIdx.x < st) red[threadIdx.x] += red[threadIdx.x + st];
        __syncthreads();
    }
    if (threadIdx.x == 0) {
        scal[S_CL] += sqrtf(red[0]);
        scal[S_SL] += sqrtf(scal[S_S2 + layer]);
    }
}

// ---------------- write outputs: p, q, s_loss, c_loss, h, d ----------------
__global__ void k_write_out(const float* __restrict__ p, const float* __restrict__ q,
                            const float* __restrict__ scal, float* __restrict__ out) {
    int i = blockIdx.x * blockDim.x + threadIdx.x;
    const int NP = N_V * PQ;
    if (i < NP)            out[i] = p[i];
    else if (i < 2 * NP)   out[i] = q[i - NP];
    else if (i == 2*NP)    out[i] = scal[S_SL];
    else if (i == 2*NP+1)  out[i] = scal[S_CL];
    else if (i == 2*NP+2)  out[i] = scal[S_POT + N_LAYERS-1] + scal[S_KIN + N_LAYERS-1];
    else if (i == 2*NP+3)  out[i] = scal[S_D + N_LAYERS-1];
}

extern "C" void kernel_launch(void* const* d_in, const int* in_sizes, int n_in,
                              void* d_out, int out_size, void* d_ws, size_t ws_size,
                              hipStream_t stream) {
    const float* vf   = (const float*)d_in[0];
    const float* ef   = (const float*)d_in[1];
    const float* mass = (const float*)d_in[2];
    const float* mvw  = (const float*)d_in[3];
    const float* vew1 = (const float*)d_in[4];
    const float* vew2 = (const float*)d_in[5];
    const float* W_e  = (const float*)d_in[6];
    const float* b_e  = (const float*)d_in[7];
    const float* W_qe = (const float*)d_in[8];
    const float* W_q  = (const float*)d_in[9];
    const float* W_p  = (const float*)d_in[10];
    const float* W_h1 = (const float*)d_in[11];
    const float* b_h1 = (const float*)d_in[12];
    const float* W_h2 = (const float*)d_in[13];
    const float* W_d1 = (const float*)d_in[14];
    const float* b_d1 = (const float*)d_in[15];
    const float* W_d2 = (const float*)d_in[16];

    float* ws = (float*)d_ws;
    size_t off = 0;
    float* uev    = ws + off; off += (size_t)N_E * UEV_DIM;
    float* B1     = ws + off; off += (size_t)UEV_DIM * B1N;
    float* bias1  = ws + off; off += B1N;
    float* HB     = ws + off; off += (size_t)N_E * B1N;
    float* DB     = ws + off; off += (size_t)N_V * HID;
    float* vq     = ws + off; off += N_V * PQ;
    float* p      = ws + off; off += N_V * PQ;
    float* q      = ws + off; off += N_V * PQ;
    float* dHdq   = ws + off; off += N_V * PQ;     // dHdq + molAcc zeroed together
    float* molAcc = ws + off; off += N_MOL * PQ;
    float* scal   = ws + off; off += 32;
    int* srcI = (int*)(ws + off); off += N_E;
    int* dstI = (int*)(ws + off); off += N_E;
    int* molI = (int*)(ws + off); off += N_V;

    const int T = 256;
    // zero the scalar slots once per launch
    k_zero<<<1, 32, 0, stream>>>(scal, 32);
    // index extraction
    k_extract_edges<<<(N_E + T - 1) / T, T, 0, stream>>>(vew1, vew2, srcI, dstI);
    k_extract_mol<<<(N_V + T - 1) / T, T, 0, stream>>>(mvw, molI);
    // uev + composite B1
    k_build_uev<<<(N_E * UEV_DIM + T - 1) / T, T, 0, stream>>>(vf, ef, srcI, dstI, uev);
    k_build_B1<<<(UEV_DIM * B1N + T - 1) / T, T, 0, stream>>>(W_h1, b_h1, W_e, b_e, B1, bias1);
    // WMMA GEMMs: HB = uev @ B1 + bias1 ;  DB = v @ W_d1[:128] + b_d1
    k_wmma_gemm_f32<UEV_DIM, B1N><<<N_E / 64, 128, 0, stream>>>(uev, B1, bias1, HB, N_E);
    k_wmma_gemm_f32<N_DIM, HID><<<N_V / 64, 128, 0, stream>>>(vf, W_d1, b_d1, DB, N_V);
    // pq encoder
    k_vertex_init<<<(N_V + T - 1) / T, T, 0, stream>>>(vf, W_qe, W_q, W_p, vq, q, p);
    k_edge_scatter_q0<<<(N_E + T - 1) / T, T, 0, stream>>>(HB, srcI, dstI, vq, q);
    // symplectic layers
    for (int layer = 0; layer < N_LAYERS; ++layer) {
        k_zero<<<(N_V * PQ + N_MOL * PQ + T - 1) / T, T, 0, stream>>>(dHdq, N_V * PQ + N_MOL * PQ);
        k_edge_layer<<<(N_E + T - 1) / T, T, 0, stream>>>(HB, srcI, dstI, q, W_h1, W_h2,
                                                          dHdq, scal, layer);
        k_vertex_layer<<<(N_V + T - 1) / T, T, 0, stream>>>(DB, mass, W_d1, W_d2, molI,
                                                            p, q, dHdq, molAcc, scal, layer);
        k_finalize_layer<<<1, 128, 0, stream>>>(molAcc, scal, layer);
    }
    // outputs
    k_write_out<<<(2 * N_V * PQ + 4 + T - 1) / T, T, 0, stream>>>(p, q, scal, (float*)d_out);
}